// motion_DF_67946382622882
// MI455X (gfx1250) — compile-verified
//
#include <hip/hip_runtime.h>
#include <hip/hip_bf16.h>

// FAC per-pixel adaptive 3x3 conv for fixed problem size N=4, C=64, H=W=256.
// Memory-bound (AI ~0.4 FLOP/B, ~704 MiB total traffic -> ~32 us at 23.3 TB/s).
// Design: one non-temporal b128 pass over motion (the 576 MiB stream),
// feature rows staged once into LDS via CDNA5 async global->LDS copies
// (GLOBAL_LOAD_ASYNC_TO_LDS_B128 + s_wait_asynccnt), non-temporal b128 stores.

#define KSZ 3
#define N_ 4
#define C_ 64
#define H_ 256
#define W_ 256
#define ROWS_PER_BLK 4
#define LROWS (ROWS_PER_BLK + 2)   // staged feature rows: h0-1 .. h0+4
#define LSTRIDE 264                 // floats per LDS row; keeps 16B segment alignment

typedef float fac_f4 __attribute__((ext_vector_type(4)));

#if defined(__HIP_DEVICE_COMPILE__)
#if __has_builtin(__builtin_amdgcn_global_load_async_to_lds_b128)
#define FAC_HAVE_ASYNC_LDS 1
#endif
#endif

#if defined(FAC_HAVE_ASYNC_LDS)
// Builtin signature (from hipcc diagnostic): param0 is
// 'int __attribute__((vector_size(16))) addrspace(1)*', param1 same in AS3,
// then two immediate ints (offset, cpol).
typedef int fac_v4i __attribute__((vector_size(16)));
typedef __attribute__((address_space(1))) fac_v4i* fac_g_v4i_p;
typedef __attribute__((address_space(3))) fac_v4i* fac_l_v4i_p;
#endif

__global__ __launch_bounds__(256)
void fac_kernel(const float* __restrict__ feature,
                const float* __restrict__ motion,
                float* __restrict__ out)
{
    const int tid = threadIdx.x;
    const int h0  = blockIdx.x * ROWS_PER_BLK;  // first output row of tile
    const int c   = blockIdx.y;
    const int n   = blockIdx.z;

    __shared__ __align__(16) float lds[LROWS * LSTRIDE];

    // Zero the halo columns (w = -1 at slot 3, w = W at slot 4+W) of each row.
    if (tid < 2 * LROWS) {
        const int row  = tid >> 1;
        const int side = tid & 1;
        lds[row * LSTRIDE + (side ? (4 + W_) : 3)] = 0.0f;
    }

    // Stage LROWS feature rows into LDS, 128 bits per task.
    // Valid data lives at lds[row*LSTRIDE + 4 + w], w in [0,W); halos zeroed.
    const size_t fplane = ((size_t)n * C_ + c) * (size_t)(H_ * W_);
    for (int idx = tid; idx < LROWS * (W_ / 4); idx += 256) {
        const int row = idx >> 6;          // W_/4 == 64 segments per row
        const int seg = idx & 63;
        const int hr  = h0 - 1 + row;
        float* dst = &lds[row * LSTRIDE + 4 + seg * 4];
        if ((unsigned)hr < (unsigned)H_) {
            const float* src = feature + fplane + (size_t)hr * W_ + seg * 4;
#if defined(FAC_HAVE_ASYNC_LDS)
            __builtin_amdgcn_global_load_async_to_lds_b128(
                (fac_g_v4i_p)src, (fac_l_v4i_p)dst,
                /*imm offset*/0, /*cpol*/0);
#else
            const fac_f4 v = *(const fac_f4*)src;
            *(fac_f4*)dst = v;
#endif
        } else {
            dst[0] = 0.0f; dst[1] = 0.0f; dst[2] = 0.0f; dst[3] = 0.0f;
        }
    }

#if defined(FAC_HAVE_ASYNC_LDS)
#if __has_builtin(__builtin_amdgcn_s_wait_asynccnt)
    __builtin_amdgcn_s_wait_asynccnt(0);
#else
    asm volatile("s_wait_asynccnt 0" ::: "memory");
#endif
#endif
    __syncthreads();

    // Each thread computes 4 consecutive pixels of one output row.
    const int r  = tid >> 6;             // row within tile (0..3)
    const int wb = (tid & 63) << 2;      // base pixel (0..252)
    const int h  = h0 + r;

    // motion[n][c*9 + t][h][wb .. wb+3], tap plane stride = H*W.
    // Read-once stream -> non-temporal loads (TH=NT) to avoid churning L2/WGP$.
    const size_t mbase =
        ((((size_t)n * C_ + c) * (KSZ * KSZ)) * H_ + h) * (size_t)W_ + wb;
    fac_f4 m[KSZ * KSZ];
#pragma unroll
    for (int t = 0; t < KSZ * KSZ; ++t)
        m[t] = __builtin_nontemporal_load(
            (const fac_f4*)(motion + mbase + (size_t)t * (H_ * W_)));

    float a0 = 0.0f, a1 = 0.0f, a2 = 0.0f, a3 = 0.0f;
#pragma unroll
    for (int i = 0; i < KSZ; ++i) {
        // lrow[k + j] == feature[h + i - 1][wb + k + j - 1] (halo = 0)
        const float* lrow = &lds[(r + i) * LSTRIDE + 3 + wb];
        const float f0 = lrow[0], f1 = lrow[1], f2 = lrow[2],
                    f3 = lrow[3], f4 = lrow[4], f5 = lrow[5];
        const fac_f4 m0 = m[i * 3 + 0];
        const fac_f4 m1 = m[i * 3 + 1];
        const fac_f4 m2 = m[i * 3 + 2];
        a0 = fmaf(f0, m0.x, a0); a0 = fmaf(f1, m1.x, a0); a0 = fmaf(f2, m2.x, a0);
        a1 = fmaf(f1, m0.y, a1); a1 = fmaf(f2, m1.y, a1); a1 = fmaf(f3, m2.y, a1);
        a2 = fmaf(f2, m0.z, a2); a2 = fmaf(f3, m1.z, a2); a2 = fmaf(f4, m2.z, a2);
        a3 = fmaf(f3, m0.w, a3); a3 = fmaf(f4, m1.w, a3); a3 = fmaf(f5, m2.w, a3);
    }

    fac_f4 o;
    o.x = a0; o.y = a1; o.z = a2; o.w = a3;
    __builtin_nontemporal_store(o, (fac_f4*)(out + fplane + (size_t)h * W_ + wb));
}

extern "C" void kernel_launch(void* const* d_in, const int* in_sizes, int n_in,
                              void* d_out, int out_size, void* d_ws, size_t ws_size,
                              hipStream_t stream) {
    (void)in_sizes; (void)n_in; (void)out_size; (void)d_ws; (void)ws_size;
    const float* feature = (const float*)d_in[0];
    const float* motion  = (const float*)d_in[1];
    float* out           = (float*)d_out;

    dim3 grid(H_ / ROWS_PER_BLK, C_, N_);
    fac_kernel<<<grid, dim3(256, 1, 1), 0, stream>>>(feature, motion, out);
}